// FocusedLinearAttention_37915971289625
// MI455X (gfx1250) — compile-verified
//
#include <hip/hip_runtime.h>

#define CDIM  64
#define KSZ   5
#define EPSV  1e-6f
#define SP    66   // padded LDS row stride in floats (even -> 8B-aligned k-pairs)

typedef float v2f __attribute__((ext_vector_type(2)));
typedef float v4f __attribute__((ext_vector_type(4)));
typedef float v8f __attribute__((ext_vector_type(8)));

__device__ __forceinline__ v8f wmma_f32_4(v2f a, v2f b, v8f c) {
  // D(16x16,f32) = A(16x4,f32) x B(4x16,f32) + C
  return __builtin_amdgcn_wmma_f32_16x16x4_f32(false, a, false, b, (short)0, c, false, false);
}

__global__ void __launch_bounds__(256, 1)
fla_fused_kernel(const float* __restrict__ x,
                 const float* __restrict__ qkv_w,
                 const float* __restrict__ qkv_b,
                 const float* __restrict__ proj_w,
                 const float* __restrict__ proj_b,
                 const float* __restrict__ scale,
                 const float* __restrict__ pos_enc,
                 const float* __restrict__ dwc_w,
                 const float* __restrict__ dwc_b,
                 float* __restrict__ out)
{
  extern __shared__ float smem[];
  // Wq: qkv_w^T in k-pair layout: Wq[(k>>1)*384 + n*2 + (k&1)]  (64x192 floats)
  float* Wq    = smem;             // 12288
  // Wp: proj_w^T in k-pair layout: Wp[(k>>1)*128 + n*2 + (k&1)] (64x64 floats)
  float* Wp    = Wq + 64*192;      // 4096
  float* xw    = Wp + 64*64;       // [64][SP] window tile; reused as attn+dwc output
  float* qb    = xw + 64*SP;       // [64][SP]
  float* kb    = qb + 64*SP;       // [64][SP]
  float* vb    = kb + 64*SP;       // [64][SP]
  float* bqkv  = vb + 64*SP;       // 192
  float* bp    = bqkv + 192;       // 64
  float* scb   = bp + 64;          // 64  softplus(scale)
  float* dwcw  = scb + 64;         // 200
  float* dwcb  = dwcw + 200;       // 8
  float* kmean = dwcb + 8;         // 64   [h][d]
  float* kvb   = kmean + 64;       // 512  [h][d][e]
  float* zb    = kvb + 512;        // 512  [n][h]

  const int tid  = threadIdx.x;
  const int lane = tid & 31;
  const int wv   = tid >> 5;       // wave 0..7
  const int l15  = lane & 15;
  const int hi   = lane >> 4;      // 0 or 1 (half-wave)

  const int w  = blockIdx.x;       // 8192 windows
  const int b  = w >> 10;
  const int wy = (w >> 5) & 31;
  const int wx = w & 31;

  // ---------------- load weights (k-pair transposed) + window tile ----------------
  for (int idx = tid; idx < 64*192; idx += 256) {
    int n = idx >> 6, k = idx & 63;
    Wq[(k >> 1)*384 + n*2 + (k & 1)] = qkv_w[idx];
  }
  for (int idx = tid; idx < 64*64; idx += 256) {
    int n = idx >> 6, k = idx & 63;
    Wp[(k >> 1)*128 + n*2 + (k & 1)] = proj_w[idx];
  }
  if (tid < 192) bqkv[tid] = qkv_b[tid];
  if (tid < 64) {
    bp[tid]  = proj_b[tid];
    scb[tid] = log1pf(expf(scale[tid]));  // softplus
  }
  if (tid < 200) dwcw[tid] = dwc_w[tid];
  if (tid < 8)   dwcb[tid] = dwc_b[tid];

  for (int s = tid; s < 512; s += 256) {          // s = c*8 + i
    int c = s >> 3, i = s & 7;
    const v4f* gp = (const v4f*)(x + (((size_t)(b*CDIM + c)*256) + (size_t)(wy*8 + i))*256 + wx*8);
    v4f r0 = __builtin_nontemporal_load(gp);
    v4f r1 = __builtin_nontemporal_load(gp + 1);
    float* lp = xw + (i*8)*SP + c;                // xw[n=i*8+j][c]
    lp[0*SP] = r0.x; lp[1*SP] = r0.y; lp[2*SP] = r0.z; lp[3*SP] = r0.w;
    lp[4*SP] = r1.x; lp[5*SP] = r1.y; lp[6*SP] = r1.z; lp[7*SP] = r1.w;
  }
  __syncthreads();

  // ---------------- GEMM1: qkv(64x192) = xw(64x64) @ qkv_w^T + bias ----------------
#pragma unroll 1
  for (int t = 0; t < 6; ++t) {
    int tile = wv*6 + t;                 // 48 tiles of 16x16
    int tm = tile / 12, tn = tile % 12;
    int col = tn*16 + l15;
    const float* arow = xw + (tm*16 + l15)*SP + hi*2;   // 8B aligned (SP even)
    const v2f*   bP   = (const v2f*)Wq + hi*192 + col;  // pair (k0,k0+1) contiguous
    v8f acc = {0,0,0,0,0,0,0,0};
#pragma unroll
    for (int ks = 0; ks < 16; ++ks) {
      v2f a  = *(const v2f*)(arow + ks*4);   // A[m][k0], A[m][k0+1]
      v2f bv = bP[ks*384];                   // W^T[k0][n], W^T[k0+1][n]
      acc = wmma_f32_4(a, bv, acc);
    }
    float bias = bqkv[col];
    float* dst; int c2;
    if      (col < 64)  { dst = qb; c2 = col; }
    else if (col < 128) { dst = kb; c2 = col - 64; }
    else                { dst = vb; c2 = col - 128; }
#pragma unroll
    for (int r = 0; r < 8; ++r) {        // C/D layout: VGPR r = rows r / r+8
      int row = tm*16 + hi*8 + r;
      dst[row*SP + c2] = acc[r] + bias;
    }
  }
  __syncthreads();

  // ---------------- focusing: relu+eps, /softplus, cube + norm rescale ----------------
  if (tid < 128) {
    const int isK = tid >> 6;
    const int n   = tid & 63;
    float* rowp = (isK ? kb : qb) + n*SP;
    const float* pe = pos_enc + n*CDIM;
    float s2 = 0.f, s6 = 0.f;
    for (int c = 0; c < CDIM; ++c) {
      float v = rowp[c];
      if (isK) v += pe[c];
      v = fmaxf(v, 0.f) + EPSV;
      v = v / scb[c];
      rowp[c] = v;
      float v2 = v*v;
      s2 += v2;
      s6 += v2*v2*v2;
    }
    float f = sqrtf(s2) * rsqrtf(s6);     // ||q|| / ||q^3||
    for (int c = 0; c < CDIM; ++c) {
      float v = rowp[c];
      rowp[c] = v*v*v*f;
    }
  }
  __syncthreads();

  // ---------------- kmean (per head-dim) and kv = (k s)^T (v s), s^2 = 1/64 ----------------
  if (tid < 64) {
    int h = tid >> 3, d = tid & 7;
    float s = 0.f;
    for (int n = 0; n < 64; ++n) s += kb[n*SP + h*8 + d];
    kmean[tid] = s * (1.f/64.f);
  }
#pragma unroll
  for (int t = 0; t < 2; ++t) {
    int idx = tid*2 + t;                 // h*64 + d*8 + e
    int h = idx >> 6, d = (idx >> 3) & 7, e = idx & 7;
    float s = 0.f;
    for (int n = 0; n < 64; ++n)
      s += kb[n*SP + h*8 + d] * vb[n*SP + h*8 + e];
    kvb[idx] = s * (1.f/64.f);
  }
  __syncthreads();

  // ---------------- z = 1/(q . kmean + eps) ----------------
#pragma unroll
  for (int t = 0; t < 2; ++t) {
    int idx = tid*2 + t;                 // n*8 + h
    int n = idx >> 3, h = idx & 7;
    float s = 0.f;
#pragma unroll
    for (int d = 0; d < 8; ++d) s += qb[n*SP + h*8 + d] * kmean[h*8 + d];
    zb[idx] = 1.f / (s + EPSV);
  }
  __syncthreads();

  // ---------------- out = (q @ kv) * z + depthwise-conv(v), into xw (reused) ----------------
  float* ob = xw;
#pragma unroll 1
  for (int t = 0; t < 16; ++t) {
    int idx = tid + t*256;               // n*64 + c
    int n = idx >> 6, c = idx & 63;
    int h = c >> 3, e = c & 7;
    float s = 0.f;
#pragma unroll
    for (int d = 0; d < 8; ++d)
      s += qb[n*SP + h*8 + d] * kvb[h*64 + d*8 + e];
    float val = s * zb[n*8 + h];
    int i = n >> 3, j = n & 7;
    float a = dwcb[e];
#pragma unroll
    for (int u = 0; u < KSZ; ++u) {
      int iy = i + u - 2;
      if (iy < 0 || iy > 7) continue;
#pragma unroll
      for (int vv = 0; vv < KSZ; ++vv) {
        int jx = j + vv - 2;
        if (jx < 0 || jx > 7) continue;
        a += vb[(iy*8 + jx)*SP + c] * dwcw[e*25 + u*5 + vv];
      }
    }
    ob[n*SP + c] = val + a;
  }
  __syncthreads();

  // ---------------- GEMM2: final(64x64) = o @ proj_w^T + bias, into qb (reused) ----------------
#pragma unroll 1
  for (int t = 0; t < 2; ++t) {
    int tile = wv*2 + t;                 // 16 tiles
    int tm = tile >> 2, tn = tile & 3;
    int col = tn*16 + l15;
    const float* arow = ob + (tm*16 + l15)*SP + hi*2;
    const v2f*   bP   = (const v2f*)Wp + hi*64 + col;
    v8f acc = {0,0,0,0,0,0,0,0};
#pragma unroll
    for (int ks = 0; ks < 16; ++ks) {
      v2f a  = *(const v2f*)(arow + ks*4);
      v2f bv = bP[ks*128];
      acc = wmma_f32_4(a, bv, acc);
    }
    float bias = bp[col];
#pragma unroll
    for (int r = 0; r < 8; ++r) {
      int row = tm*16 + hi*8 + r;
      qb[row*SP + col] = acc[r] + bias;
    }
  }
  __syncthreads();

  // ---------------- window-reverse scatter (nontemporal streaming stores) ----------------
  for (int s = tid; s < 512; s += 256) {
    int c = s >> 3, i = s & 7;
    v4f* gp = (v4f*)(out + (((size_t)(b*CDIM + c)*256) + (size_t)(wy*8 + i))*256 + wx*8);
    const float* lp = qb + (i*8)*SP + c;
    v4f w0, w1;
    w0.x = lp[0*SP]; w0.y = lp[1*SP]; w0.z = lp[2*SP]; w0.w = lp[3*SP];
    w1.x = lp[4*SP]; w1.y = lp[5*SP]; w1.z = lp[6*SP]; w1.w = lp[7*SP];
    __builtin_nontemporal_store(w0, gp);
    __builtin_nontemporal_store(w1, gp + 1);
  }
}

extern "C" void kernel_launch(void* const* d_in, const int* in_sizes, int n_in,
                              void* d_out, int out_size, void* d_ws, size_t ws_size,
                              hipStream_t stream) {
  const float* x      = (const float*)d_in[0];
  const float* qkv_w  = (const float*)d_in[1];
  const float* qkv_b  = (const float*)d_in[2];
  const float* proj_w = (const float*)d_in[3];
  const float* proj_b = (const float*)d_in[4];
  const float* scale  = (const float*)d_in[5];
  const float* pos_enc= (const float*)d_in[6];
  const float* dwc_w  = (const float*)d_in[7];
  const float* dwc_b  = (const float*)d_in[8];
  float* outp = (float*)d_out;

  const size_t smem_floats =
      64*192 + 64*64 + (size_t)4*64*SP + 192 + 64 + 64 + 200 + 8 + 64 + 512 + 512;
  const size_t smem = smem_floats * sizeof(float);   // ~136 KB, fits 320 KB/WGP

  (void)hipFuncSetAttribute((const void*)fla_fused_kernel,
                            hipFuncAttributeMaxDynamicSharedMemorySize, (int)smem);

  fla_fused_kernel<<<8192, 256, smem, stream>>>(
      x, qkv_w, qkv_b, proj_w, proj_b, scale, pos_enc, dwc_w, dwc_b, outp);
}